// Att_scores_66108136620087
// MI455X (gfx1250) — compile-verified
//
#include <hip/hip_runtime.h>

// ---------------------------------------------------------------------------
// scores[b,n] = X[b,n,:] . v[b,:]
//   v[b,c]   = scale * sum_j W_qkv[j,c] * ksum[b,j]          (Wq^T @ ksum)
//   ksum[b,j]= sum_c W_qkv[768+j,c] * xsum[b,c]              (Wk  @ xsum)
//   xsum[b,c]= sum_n X[b,n,c]
// Exact algebraic refactor of the reference (dropout p=0, no softmax).
// Final GEMV stage runs on V_WMMA_F32_16X16X4_F32 (fp32 WMMA, wave32).
// ---------------------------------------------------------------------------

typedef __attribute__((ext_vector_type(2))) float v2f;
typedef __attribute__((ext_vector_type(8))) float v8f;

#define BATCH   2
#define SEQ     2048
#define DIM     768
#define NCHUNKS 8
#define ROWS_PER_CHUNK (SEQ / NCHUNKS)   // 256

// ---- Stage 1: partial column sums of X --------------------------------------
// grid (BATCH, NCHUNKS), block 256. Fully coalesced streaming read of X (once
// from HBM; leaves X resident in the 192MB L2 for stage 3).
__global__ __launch_bounds__(256)
void colsum_partial_kernel(const float* __restrict__ X, float* __restrict__ partial) {
    const int b     = blockIdx.x;
    const int chunk = blockIdx.y;
    const int t     = threadIdx.x;
    const float* Xb = X + ((size_t)b * SEQ + (size_t)chunk * ROWS_PER_CHUNK) * DIM;
    float a0 = 0.f, a1 = 0.f, a2 = 0.f;
    for (int r = 0; r < ROWS_PER_CHUNK; ++r) {
        const float* row = Xb + (size_t)r * DIM;
        a0 += row[t];
        a1 += row[t + 256];
        a2 += row[t + 512];
    }
    float* P = partial + ((size_t)b * NCHUNKS + chunk) * DIM;
    P[t] = a0; P[t + 256] = a1; P[t + 512] = a2;
}

// ---- Stage 2: xsum -> ksum -> v (per batch, one block) ----------------------
// Tiny (2 x 768x768 matvec); W streamed once, ksum broadcast from LDS.
__global__ __launch_bounds__(256)
void build_v_kernel(const float* __restrict__ W, const float* __restrict__ partial,
                    float* __restrict__ vout) {
    __shared__ float xs[DIM];
    __shared__ float ks[DIM];
    const int b = blockIdx.x;
    const int t = threadIdx.x;

    // reduce the NCHUNKS partials into xsum
    for (int i = t; i < DIM; i += 256) {
        float s = 0.f;
        for (int c = 0; c < NCHUNKS; ++c)
            s += partial[((size_t)b * NCHUNKS + c) * DIM + i];
        xs[i] = s;
    }
    __syncthreads();

    // ksum[j] = dot(Wk[j,:], xsum) ; Wk rows = W_qkv[768 + j, :]
    for (int j = t; j < DIM; j += 256) {
        const float* wr = W + (size_t)(DIM + j) * DIM;
        float s = 0.f;
        for (int c = 0; c < DIM; ++c) s += wr[c] * xs[c];
        ks[j] = s;
    }
    __syncthreads();

    // v[c] = scale * sum_j Wq[j,c] * ksum[j]   (coalesced over c)
    const float scale = 0.125f;               // (768/12)^-0.5 = 64^-0.5
    for (int i = t; i < DIM; i += 256) {
        float s = 0.f;
        for (int j = 0; j < DIM; ++j) s += W[(size_t)j * DIM + i] * ks[j];
        vout[(size_t)b * DIM + i] = s * scale;
    }
}

// ---- Stage 3: scores = X . v via V_WMMA_F32_16X16X4_F32 ---------------------
// Block = 256 threads = 8 waves; each wave owns 16 consecutive tokens and
// accumulates D(16x16) += A(16x4 of X) * B(4x16 broadcast of v) over 192
// K-steps. ISA VGPR layouts (wave32):
//   A 16x4 f32 : lane l -> row M=l&15 ; K = 2*(l>=16) + vgpr   (v0=Klo,v1=Khi)
//   B 4x16 f32 : lane l -> col N=l&15 ; K = 2*(l>=16) + vgpr   (same K map)
//   C/D        : vgpr r -> M=r (lanes 0-15) / M=8+r (lanes 16-31), N = l&15
// Every B column equals v[k0..k0+3], so all D columns are identical; we read
// column N=0 (lanes 0 and 16). EXEC is all-ones through every WMMA.
__global__ __launch_bounds__(256)
void scores_wmma_kernel(const float* __restrict__ X, const float* __restrict__ v,
                        float* __restrict__ out) {
    __shared__ float vlds[DIM];
    const int t  = threadIdx.x;
    const int R0 = blockIdx.x * 128;          // 128 tokens per block (8 waves x 16)
    const int b  = R0 / SEQ;                  // 128 | 2048: block never straddles a batch

    for (int i = t; i < DIM; i += 256) vlds[i] = v[(size_t)b * DIM + i];
    __syncthreads();

    const int wave = t >> 5;
    const int lane = t & 31;
    const int m    = lane & 15;
    const int h2   = (lane >> 4) << 1;        // 0 for lanes 0-15, 2 for lanes 16-31
    const float* xr = X + (size_t)(R0 + wave * 16 + m) * DIM;

    v8f acc = {0.f, 0.f, 0.f, 0.f, 0.f, 0.f, 0.f, 0.f};
    for (int k0 = 0; k0 < DIM; k0 += 4) {
        const int kb = k0 + h2;
        v2f a  = *(const v2f*)(xr + kb);      // A: X[row][kb], X[row][kb+1]
        v2f bb = *(const v2f*)(vlds + kb);    // B: v[kb], v[kb+1] (bcast to all N)
        acc = __builtin_amdgcn_wmma_f32_16x16x4_f32(
                /*neg_a=*/false, a, /*neg_b=*/false, bb,
                /*c_mod=*/(short)0, acc, /*reuse_a=*/false, /*reuse_b=*/false);
    }

    // Column N=0 lives in lanes 0 (M=0..7) and 16 (M=8..15).
    if ((lane & 15) == 0) {
        const int outRow = R0 + wave * 16 + (lane >> 4) * 8;
#pragma unroll
        for (int r = 0; r < 8; ++r) out[outRow + r] = acc[r];
    }
}

// ---------------------------------------------------------------------------
extern "C" void kernel_launch(void* const* d_in, const int* in_sizes, int n_in,
                              void* d_out, int out_size, void* d_ws, size_t ws_size,
                              hipStream_t stream) {
    const float* X = (const float*)d_in[0];   // [2, 2048, 768] fp32
    const float* W = (const float*)d_in[1];   // [1536, 768]    fp32
    float* out = (float*)d_out;               // [2, 2048]      fp32

    float* partial = (float*)d_ws;                          // 2*8*768 floats
    float* vbuf    = partial + (size_t)BATCH * NCHUNKS * DIM; // 2*768 floats
    (void)in_sizes; (void)n_in; (void)out_size; (void)ws_size;

    colsum_partial_kernel<<<dim3(BATCH, NCHUNKS), 256, 0, stream>>>(X, partial);
    build_v_kernel<<<BATCH, 256, 0, stream>>>(W, partial, vbuf);
    scores_wmma_kernel<<<(BATCH * SEQ) / 128, 256, 0, stream>>>(X, vbuf, out);
}